// TTEmbedding_3255585210388
// MI455X (gfx1250) — compile-verified
//
#include <hip/hip_runtime.h>

// TT-embedding: out[n,1024] = (g0[i0] (8x8) @ g1[i1] (8x64)) viewed (64x8) @ (mask * g2[i2] (8x16))
// VOC_QUANT=(32,32,32), EMB_QUANT=(8,8,16), RANKS=(1,8,8,1), PADDING_IDX=0.
// Both contractions on v_wmma_f32_16x16x4_f32; core1 staged to LDS via the Tensor Data Mover.

typedef float v2f __attribute__((ext_vector_type(2)));
typedef float v4f __attribute__((ext_vector_type(4)));
typedef float v8f __attribute__((ext_vector_type(8)));
typedef unsigned int u32x4 __attribute__((ext_vector_type(4)));
typedef int i32x4 __attribute__((ext_vector_type(4)));
typedef int i32x8 __attribute__((ext_vector_type(8)));

#define WAVES 8
#define BLOCK (WAVES * 32)

#if __has_builtin(__builtin_amdgcn_tensor_load_to_lds) && \
    __has_builtin(__builtin_amdgcn_s_wait_tensorcnt)
#define USE_TDM 1
#else
#define USE_TDM 0
#endif

__device__ __forceinline__ v8f wmma4(v2f a, v2f b, v8f c) {
  // 8 args: (neg_a, A, neg_b, B, c_mod, C, reuse_a, reuse_b)
  return __builtin_amdgcn_wmma_f32_16x16x4_f32(false, a, false, b, (short)0, c,
                                               false, false);
}

__global__ __launch_bounds__(BLOCK) void tt_embedding_wmma_kernel(
    const int* __restrict__ x,        // [N] flat vocab indices
    const float* __restrict__ core0,  // [32][1][8][8]  -> per i0: 8x8  (e0 x r1)
    const float* __restrict__ core1,  // [32][8][8][8]  -> per i1: 8x64 (r1 x (e1*8+r2))
    const float* __restrict__ core2,  // [32][8][16][1] -> per i2: 8x16 (r2 x e2)
    float* __restrict__ out,          // [N][1024]
    int N) {
  __shared__ __align__(16) float lds_c0[2048];            // 8 KB
  __shared__ __align__(16) float lds_c1[16384];           // 64 KB
  __shared__ __align__(16) float lds_c2[4096];            // 16 KB
  __shared__ __align__(16) float lds_res1[WAVES * 1024];  // 32 KB per-wave bounce

  const int tid = threadIdx.x;

  // ---- Stage cores into LDS -----------------------------------------------
  {
    v4f* s = reinterpret_cast<v4f*>(lds_c0);
    const v4f* g = reinterpret_cast<const v4f*>(core0);
    for (int i = tid; i < 2048 / 4; i += BLOCK) s[i] = g[i];
  }
  {
    v4f* s = reinterpret_cast<v4f*>(lds_c2);
    const v4f* g = reinterpret_cast<const v4f*>(core2);
    for (int i = tid; i < 4096 / 4; i += BLOCK) s[i] = g[i];
  }
#if USE_TDM
  // core1 (64 KB) via Tensor Data Mover: single 1x16384-element tile, 4B elems.
  if (tid < 32) {
    const unsigned long long ga = (unsigned long long)(uintptr_t)core1;
    // LDS flat address truncates to the in-workgroup LDS byte offset (ISA 10.2).
    const unsigned int ldsb = (unsigned int)(uintptr_t)(void*)lds_c1;
    u32x4 d0{};
    d0[0] = 1u;                      // count=1, is_restore=0, gather off
    d0[1] = ldsb;                    // lds_addr  (bits 63:32)
    d0[2] = (unsigned int)ga;        // global_addr[31:0]
    d0[3] = (unsigned int)((ga >> 32) & 0x01FFFFFFu) | (2u << 30);  // type=2
    i32x8 d1{};
    d1[0] = (2 << 16);               // wg_mask=0, data_size=2 (4 bytes)
    d1[1] = (int)((16384u & 0xFFFFu) << 16);  // tensor_dim0 lo16 @ [63:48]
    d1[2] = (int)(((16384u >> 16) & 0xFFFFu) | (1u << 16));  // dim0 hi | dim1 lo
    d1[3] = (int)((16384u & 0xFFFFu) << 16);  // dim1 hi=0 | tile_dim0=16384
    d1[4] = 1;                       // tile_dim1=1, tile_dim2=0
    d1[5] = 16384;                   // tensor_dim0_stride lo32
    d1[6] = 0;                       // stride0 hi | stride1 lo
    d1[7] = 0;
    i32x4 d2{};
    i32x4 d3{};
    i32x8 d4{};                      // unused trailing group (6-arg toolchain form)
    __builtin_amdgcn_tensor_load_to_lds(d0, d1, d2, d3, d4, 0);
    __builtin_amdgcn_s_wait_tensorcnt((short)0);
  }
#else
  {
    v4f* s = reinterpret_cast<v4f*>(lds_c1);
    const v4f* g = reinterpret_cast<const v4f*>(core1);
    for (int i = tid; i < 16384 / 4; i += BLOCK) s[i] = g[i];
  }
#endif
  __syncthreads();

  const int lane = tid & 31;
  const int wave = tid >> 5;
  const int half = lane >> 4;  // 0: lanes 0-15, 1: lanes 16-31
  const int l15 = lane & 15;
  float* r1buf = lds_res1 + wave * 1024;

  for (int n = blockIdx.x * WAVES + wave; n < N; n += (int)gridDim.x * WAVES) {
    const int flat = x[n];
    const int i0 = (flat >> 10) & 31;
    const int i1 = (flat >> 5) & 31;
    const int i2 = flat & 31;
    const float* g0 = lds_c0 + i0 * 64;   // [e0][r1]        8x8
    const float* g1 = lds_c1 + i1 * 512;  // [r1][e1*8+r2]   8x64
    const float* g2 = lds_c2 + i2 * 128;  // [r2][e2]        8x16
    const float mask = (flat != 0) ? 1.0f : 0.0f;  // folded into stage-2 B

    // ============ Stage 1: (8x8) @ (8x64) -> 16x64 (rows 8..15 junk) =======
    // A 16x4 f32: lanes 0-15 row M=l15 {K=k0,k0+1}; lanes 16-31 {K=k0+2,k0+3}
    // B  4x16 f32: lanes 0-15 {K=k0,k0+1} @ N=l15; lanes 16-31 {K=k0+2,k0+3}
    v8f acc1[4] = {v8f{}, v8f{}, v8f{}, v8f{}};
#pragma unroll
    for (int c = 0; c < 2; ++c) {  // K=8 -> two K=4 chunks
      const int k0 = c * 4 + half * 2;
      v2f a;
      if (l15 < 8) {
        a.x = g0[l15 * 8 + k0];
        a.y = g0[l15 * 8 + k0 + 1];
      } else {
        a.x = 0.0f;
        a.y = 0.0f;
      }
#pragma unroll
      for (int t = 0; t < 4; ++t) {  // 4 N-tiles over 64 columns
        const int col = t * 16 + l15;
        v2f b;
        b.x = g1[k0 * 64 + col];
        b.y = g1[(k0 + 1) * 64 + col];
        acc1[t] = wmma4(a, b, acc1[t]);
      }
    }
    // D-layout -> A-layout via LDS. Valid rows (e0=0..7) land in r1buf[0..511].
#pragma unroll
    for (int t = 0; t < 4; ++t) {
#pragma unroll
      for (int v = 0; v < 8; ++v) {
        r1buf[(v + half * 8) * 64 + t * 16 + l15] = acc1[t][v];
      }
    }

    // ============ Stage 2: res1 (64x8) @ (mask*g2) (8x16) -> 64x16 =========
    v8f acc2[4] = {v8f{}, v8f{}, v8f{}, v8f{}};
#pragma unroll
    for (int c = 0; c < 2; ++c) {
      const int k0 = c * 4 + half * 2;
      v2f b;
      b.x = g2[k0 * 16 + l15] * mask;
      b.y = g2[(k0 + 1) * 16 + l15] * mask;
#pragma unroll
      for (int t = 0; t < 4; ++t) {  // 4 M-tiles over 64 rows
        const int m = t * 16 + l15;
        v2f a;
        a.x = r1buf[m * 8 + k0];
        a.y = r1buf[m * 8 + k0 + 1];
        acc2[t] = wmma4(a, b, acc2[t]);
      }
    }

    // ============ Coalesced streaming store via LDS bounce ==================
    // r1buf (stage-1 data now dead) <- res2 row-major [64][16], then each lane
    // streams 8 x b128 contiguous chunks (512B per instruction across the wave).
#pragma unroll
    for (int t = 0; t < 4; ++t) {
#pragma unroll
      for (int v = 0; v < 8; ++v) {
        r1buf[(t * 16 + v + half * 8) * 16 + l15] = acc2[t][v];
      }
    }
    float* obase = out + (size_t)n * 1024;
#pragma unroll
    for (int c = 0; c < 8; ++c) {
      const int off = c * 128 + lane * 4;
      v4f vv = *reinterpret_cast<const v4f*>(r1buf + off);
      __builtin_nontemporal_store(vv, reinterpret_cast<v4f*>(obase + off));
    }
  }
}

extern "C" void kernel_launch(void* const* d_in, const int* in_sizes, int n_in,
                              void* d_out, int out_size, void* d_ws,
                              size_t ws_size, hipStream_t stream) {
  const int* x = (const int*)d_in[0];
  const float* core0 = (const float*)d_in[1];
  const float* core1 = (const float*)d_in[2];
  const float* core2 = (const float*)d_in[3];
  float* out = (float*)d_out;
  const int N = in_sizes[0];

  int blocks = (N + WAVES - 1) / WAVES;
  if (blocks > 512) blocks = 512;  // 512 x 88KB preload ~ 45MB << 128MB output
  if (blocks < 1) blocks = 1;

  tt_embedding_wmma_kernel<<<blocks, BLOCK, 0, stream>>>(x, core0, core1,
                                                         core2, out, N);
}